// GCN_56100862820624
// MI455X (gfx1250) — compile-verified
//
#include <hip/hip_runtime.h>

#define NN   100000
#define EE   1600000
#define GG   512
#define FIN  128
#define FHID 128
#define FOUT 64

typedef float v2f __attribute__((ext_vector_type(2)));
typedef float v8f __attribute__((ext_vector_type(8)));

// ---------------- degree / normalization ----------------

__global__ void deg_init_kernel(float* deg, int n) {
  int i = blockIdx.x * blockDim.x + threadIdx.x;
  if (i < n) deg[i] = 1.0f;                    // self-loop
}

__global__ void deg_count_kernel(const int* __restrict__ dst, float* deg, int e) {
  int i = blockIdx.x * blockDim.x + threadIdx.x;
  if (i < e) atomicAdd(&deg[dst[i]], 1.0f);
}

__global__ void deg_rsqrt_kernel(float* deg, int n) {
  int i = blockIdx.x * blockDim.x + threadIdx.x;
  if (i < n) deg[i] = rsqrtf(deg[i]);          // deg -> dinv in place
}

// ---------------- fp32 WMMA GEMM with LDS-staged A band -----------------------
// C[M x Nc] = A[M x K] @ B[K x Nc] (+bias).  One block per 16-row band of A
// (grid = M/16), blockDim = 32 * (Nc/16): wave w computes the 16x16 tile at
// column band w.  The 16 x K A band is staged once in LDS (coalesced float4
// fill), row stride padded to K+4 floats so the 16 rows read per WMMA step hit
// disjoint LDS banks.  K stepped by 4 via V_WMMA_F32_16X16X4_F32.
// A 16x4 layout: lanes 0-15 -> M=lane, (K=k,k+1); lanes 16-31 -> M=lane-16, (K=k+2,k+3)
// B 4x16 layout: lanes 0-15 -> N=lane, (K=k,k+1); lanes 16-31 -> N=lane-16, (K=k+2,k+3)
// C/D: vgpr r, lane L -> row = r + 8*(L>=16), col = L%16

__global__ void gemm_wmma_f32_kernel(const float* __restrict__ A,
                                     const float* __restrict__ B,
                                     const float* __restrict__ bias,
                                     float* __restrict__ C,
                                     int Nc, int K) {
  __shared__ float As[16 * (128 + 4)];         // K <= 128, padded rows

  int tM   = blockIdx.x;
  int wave = threadIdx.x >> 5;
  int lane = threadIdx.x & 31;
  int ldsK = K + 4;

  // cooperative, coalesced stage of the 16 x K band into LDS
  int K4 = K >> 2;
  for (int i = threadIdx.x; i < 16 * K4; i += blockDim.x) {
    int r = i / K4;
    int c = i - r * K4;
    ((float4*)(As + r * ldsK))[c] =
        ((const float4*)(A + ((size_t)(tM * 16 + r)) * K))[c];
  }
  __syncthreads();

  int half = lane >> 4;                        // 0: lanes 0-15, 1: lanes 16-31
  int l    = lane & 15;
  int bcol = wave * 16 + l;

  const float* ap = As + l * ldsK + half * 2;  // A row l of the band (in LDS)
  v8f acc = {};

  for (int k = 0; k < K; k += 4) {
    v2f a;
    a.x = ap[k];
    a.y = ap[k + 1];
    int kb = k + half * 2;
    v2f b;
    b.x = B[(size_t)kb * Nc + bcol];
    b.y = B[(size_t)(kb + 1) * Nc + bcol];
    acc = __builtin_amdgcn_wmma_f32_16x16x4_f32(
        /*neg_a=*/false, a, /*neg_b=*/false, b,
        /*c_mod=*/(short)0, acc, /*reuse_a=*/false, /*reuse_b=*/false);
  }

  float bv = bias ? bias[bcol] : 0.0f;
  int rbase = tM * 16 + half * 8;
#pragma unroll
  for (int r = 0; r < 8; ++r)
    C[(size_t)(rbase + r) * Nc + bcol] = acc[r] + bv;
}

// ---------------- aggregation ----------------

// agg[i,f] = hw[i,f] * dinv[i]^2 (self-loop, norm = dinv_i*dinv_i) + bias[f]
__global__ void agg_init_kernel(const float* __restrict__ hw,
                                const float* __restrict__ dinv,
                                const float* __restrict__ bias,
                                float* __restrict__ agg, int n, int F) {
  int idx = blockIdx.x * blockDim.x + threadIdx.x;
  if (idx >= n * F) return;
  int i = idx / F;
  int f = idx - i * F;
  float di = dinv[i];
  agg[idx] = hw[idx] * di * di + bias[f];
}

// one wave per edge: gather hw[src] (float4/lane), scale by dinv_s*dinv_d,
// scatter-add into agg[dst] with fp32 atomics (L2-resident working set)
__global__ void agg_edge_kernel(const float* __restrict__ hw,
                                const float* __restrict__ dinv,
                                const int* __restrict__ src,
                                const int* __restrict__ dst,
                                float* __restrict__ agg, int e) {
  int w = (int)((blockIdx.x * blockDim.x + threadIdx.x) >> 5);
  if (w >= e) return;
  int lane = threadIdx.x & 31;
  int s = src[w], d = dst[w];
  float norm = dinv[s] * dinv[d];
  float4 v = ((const float4*)(hw + (size_t)s * FHID))[lane];
  float* p = agg + (size_t)d * FHID + lane * 4;
  atomicAdd(p + 0, v.x * norm);
  atomicAdd(p + 1, v.y * norm);
  atomicAdd(p + 2, v.z * norm);
  atomicAdd(p + 3, v.w * norm);
}

__global__ void relu_kernel(float* x, int n) {
  int i = blockIdx.x * blockDim.x + threadIdx.x;
  if (i < n) x[i] = fmaxf(x[i], 0.0f);
}

// ---------------- global mean pool ----------------

__global__ void pool_zero_kernel(float* pooled, float* cnt, int gF, int g) {
  int i = blockIdx.x * blockDim.x + threadIdx.x;
  if (i < gF) pooled[i] = 0.0f;
  if (i < g) cnt[i] = 0.0f;
}

__global__ void pool_count_kernel(const int* __restrict__ batch, float* cnt, int n) {
  int i = blockIdx.x * blockDim.x + threadIdx.x;
  if (i < n) atomicAdd(&cnt[batch[i]], 1.0f);
}

__global__ void pool_sum_kernel(const float* __restrict__ h,
                                const int* __restrict__ batch,
                                float* __restrict__ pooled, int n) {
  int w = (int)((blockIdx.x * blockDim.x + threadIdx.x) >> 5);
  if (w >= n) return;
  int lane = threadIdx.x & 31;
  int g = batch[w];
  float4 v = ((const float4*)(h + (size_t)w * FHID))[lane];
  float* p = pooled + (size_t)g * FHID + lane * 4;
  atomicAdd(p + 0, v.x);
  atomicAdd(p + 1, v.y);
  atomicAdd(p + 2, v.z);
  atomicAdd(p + 3, v.w);
}

__global__ void pool_div_kernel(float* pooled, const float* __restrict__ cnt, int gF) {
  int i = blockIdx.x * blockDim.x + threadIdx.x;
  if (i < gF) pooled[i] *= 1.0f / fmaxf(cnt[i >> 7], 1.0f);   // FHID == 128
}

// ---------------- launch ----------------

extern "C" void kernel_launch(void* const* d_in, const int* in_sizes, int n_in,
                              void* d_out, int out_size, void* d_ws, size_t ws_size,
                              hipStream_t stream) {
  const float* x    = (const float*)d_in[0];
  const int*   ei   = (const int*)d_in[1];      // [2, E] flat
  const int*   batch= (const int*)d_in[3];
  const float* W1   = (const float*)d_in[4];
  const float* b1   = (const float*)d_in[5];
  const float* W2   = (const float*)d_in[6];
  const float* b2   = (const float*)d_in[7];
  const float* Wfc  = (const float*)d_in[8];
  const float* bfc  = (const float*)d_in[9];
  float* out = (float*)d_out;

  const int* src = ei;
  const int* dst = ei + EE;

  // workspace layout (~103 MB)
  char* ws = (char*)d_ws;
  float* hw     = (float*)(ws);                                          // N*128
  float* agg    = (float*)(ws + (size_t)NN * FHID * 4);                  // N*128
  float* dinv   = (float*)(ws + (size_t)NN * FHID * 8);                  // N
  float* pooled = (float*)(ws + (size_t)NN * FHID * 8 + (size_t)NN * 4); // G*128
  float* cnt    = pooled + (size_t)GG * FHID;                            // G

  dim3 blk(256);
  int nfBlocks   = (NN * FHID + 255) / 256;       // 50000
  int edgeBlocks = (EE * 32 + 255) / 256;         // 200000 (wave per edge)

  // symmetric normalization (shared by both convs)
  deg_init_kernel <<<(NN + 255) / 256, blk, 0, stream>>>(dinv, NN);
  deg_count_kernel<<<(EE + 255) / 256, blk, 0, stream>>>(dst, dinv, EE);
  deg_rsqrt_kernel<<<(NN + 255) / 256, blk, 0, stream>>>(dinv, NN);

  // GEMM launch shapes: grid = M/16 bands, blockDim = 32 * (Nc/16)
  dim3 gemmBlk(32 * (FHID / 16));                 // 256 threads, 8 column tiles
  int  gemmGrid = NN / 16;                        // 6250

  // conv1: hw = x @ W1 ; agg = norm-aggregate + b1 ; ReLU
  gemm_wmma_f32_kernel<<<gemmGrid, gemmBlk, 0, stream>>>(x, W1, nullptr, hw, FHID, FIN);
  agg_init_kernel<<<nfBlocks, blk, 0, stream>>>(hw, dinv, b1, agg, NN, FHID);
  agg_edge_kernel<<<edgeBlocks, blk, 0, stream>>>(hw, dinv, src, dst, agg, EE);
  relu_kernel<<<nfBlocks, blk, 0, stream>>>(agg, NN * FHID);

  // conv2: hw = h1 @ W2 ; agg = norm-aggregate + b2
  gemm_wmma_f32_kernel<<<gemmGrid, gemmBlk, 0, stream>>>(agg, W2, nullptr, hw, FHID, FHID);
  agg_init_kernel<<<nfBlocks, blk, 0, stream>>>(hw, dinv, b2, agg, NN, FHID);
  agg_edge_kernel<<<edgeBlocks, blk, 0, stream>>>(hw, dinv, src, dst, agg, EE);

  // global mean pool
  pool_zero_kernel <<<(GG * FHID + 255) / 256, blk, 0, stream>>>(pooled, cnt, GG * FHID, GG);
  pool_count_kernel<<<(NN + 255) / 256, blk, 0, stream>>>(batch, cnt, NN);
  pool_sum_kernel  <<<(NN * 32 + 255) / 256, blk, 0, stream>>>(agg, batch, pooled, NN);
  pool_div_kernel  <<<(GG * FHID + 255) / 256, blk, 0, stream>>>(pooled, cnt, GG * FHID);

  // final FC: out = pooled @ Wfc + bfc   (512x128x64, bias fused)
  dim3 fcBlk(32 * (FOUT / 16));                   // 128 threads, 4 column tiles
  gemm_wmma_f32_kernel<<<GG / 16, fcBlk, 0, stream>>>(pooled, Wfc, bfc, out, FOUT, FHID);
}